// Gemma4AudioRelativePositionEmbedding_20727512170990
// MI455X (gfx1250) — compile-verified
//
#include <hip/hip_runtime.h>

// ---------------------------------------------------------------------------
// Gemma4 audio relative position embedding, MI455X (gfx1250).
//
// Roofline: 225 MB traffic @ 23.3 TB/s => ~9.7us; AI ~26 FLOP/B < f32-matrix
// ridge, so the kernel is memory-bound even in full f32 => use
// V_WMMA_F32_16X16X4_F32 (no precision downconversion needed).
//
// sin_emb shortcut: inv_timescales == 1 makes every sinusoid row constant,
// so sin_emb[f,n,h] = sin(pos_f)*S1[n*128+h] + cos(pos_f)*S2[n*128+h] with
// S1/S2 = column sums of the two halves of pos_proj (tiny kernel 1).
//
// relative_shift algebra: out[w,c] = ac[w,c] + (0 <= c-w < 24 ? bd[w,c-w] : 0).
//
// Wave tiling: 8 waves = 4 M-groups x 2 N-groups. Each wave owns a 2Mx6N'
// block (N' = 10 C-tiles + 2 sin_emb tiles appended to the same LDS region),
// so the inner loop is 8 ds_load_b64 : 12 v_wmma (B fragments shared across
// both M-tiles) instead of 13 : 12 when every wave spans the full N range.
// ---------------------------------------------------------------------------

#define NUM_HEADS 8
#define HEAD_DIM 128
#define HIDDEN 1024
#define MAX_BACK 11
#define SPAN 24
#define B_ 8
#define U_ 16
#define W_ 128
#define C_ 151
#define CPAD 160            // 10 C-tiles of 16
#define FPAD 32             // 2 sin_emb tiles of 16
#define NPROWS (CPAD + FPAD) // 192 unified B rows (12 N'-tiles)
#define LSTR 132   // LDS row stride (floats): 132 mod 64 = 4 -> conflict-free
                   // b64 fragment reads (lanes 0-15 and 16-31 hit disjoint banks)
#define BDSTR 25   // odd stride for the bd scratch

typedef __attribute__((ext_vector_type(2))) float v2f;
typedef __attribute__((ext_vector_type(8))) float v8f;

// ---------------------------------------------------------------------------
// Kernel 1: collapse the 24x1024 @ 1024x1024 sinusoid projection into two
// column-sum reductions, write semb[n][f(FPAD)][h] (zeros for f >= SPAN).
// ---------------------------------------------------------------------------
__global__ void semb_kernel(const float* __restrict__ pos_proj,
                            float* __restrict__ semb) {
  const int j = blockIdx.x * blockDim.x + threadIdx.x;  // 0..1023
  float s1 = 0.f, s2 = 0.f;
  for (int i = 0; i < HIDDEN / 2; ++i)      s1 += pos_proj[i * HIDDEN + j];
  for (int i = HIDDEN / 2; i < HIDDEN; ++i) s2 += pos_proj[i * HIDDEN + j];
  const int n = j >> 7, h = j & 127;
  for (int f = 0; f < FPAD; ++f) {
    float v = 0.f;
    if (f < SPAN) {
      const float pos = (float)(MAX_BACK - f);
      v = sinf(pos) * s1 + cosf(pos) * s2;
    }
    semb[(size_t)(n * FPAD + f) * HEAD_DIM + h] = v;
  }
}

// ---------------------------------------------------------------------------
// Kernel 2: one workgroup (8 waves) per (b, n, u).
// ---------------------------------------------------------------------------
__global__ void __launch_bounds__(256, 1)
attn_kernel(const float* __restrict__ q, const float* __restrict__ k,
            const float* __restrict__ semb, float* __restrict__ out) {
  __shared__ float Alds[W_ * LSTR];       //  67.6 KB  q tile   [w][h]
  __shared__ float BSlds[NPROWS * LSTR];  // 101.4 KB  k tile   [c][h] rows 0..159
                                          //           sin_emb  [f][h] rows 160..191
  __shared__ float BDlds[W_ * BDSTR];     //  12.8 KB  bd scratch for rel-shift

  const int blk = blockIdx.x;             // ((b*8 + n)*16 + u)
  const int u = blk & 15;
  const int n = (blk >> 4) & 7;
  const int b = blk >> 7;

  const int tid  = threadIdx.x;
  const int lane = tid & 31;
  const int wv   = tid >> 5;
  const int mgrp = wv >> 1;               // 0..3 -> M-tiles {2*mgrp, 2*mgrp+1}
  const int ngrp = wv & 1;                // 0..1 -> N'-tiles [6*ngrp, 6*ngrp+6)
  const int half = lane >> 4;             // 0: K+0/K+1 lanes, 1: K+2/K+3 lanes
  const int l16  = lane & 15;

  const size_t qbase = (size_t)(b * U_ + u) * W_ * HIDDEN + n * HEAD_DIM;
  const size_t kbase = (size_t)(b * U_ + u) * C_ * HIDDEN + n * HEAD_DIM;

  // Stage A: queries[b,u,w,n,:] -> Alds[w][h]  (float4, coalesced)
  #pragma unroll
  for (int i = 0; i < (W_ * HEAD_DIM / 4) / 256; ++i) {       // 16 iters
    const int idx = tid + i * 256;
    const int w = idx >> 5, h4 = idx & 31;
    const float4 v = *(const float4*)(q + qbase + (size_t)w * HIDDEN + h4 * 4);
    *(float4*)&Alds[w * LSTR + h4 * 4] = v;
  }
  // Stage B rows: keys for c<151, zeros for 151<=c<160, sin_emb for c>=160.
  #pragma unroll
  for (int i = 0; i < (NPROWS * HEAD_DIM / 4) / 256; ++i) {   // 24 iters
    const int idx = tid + i * 256;
    const int c = idx >> 5, h4 = idx & 31;
    float4 v = make_float4(0.f, 0.f, 0.f, 0.f);
    if (c < C_) {
      v = *(const float4*)(k + kbase + (size_t)c * HIDDEN + h4 * 4);
    } else if (c >= CPAD) {
      v = *(const float4*)(semb + (size_t)(n * FPAD + (c - CPAD)) * HEAD_DIM +
                           h4 * 4);
    }
    *(float4*)&BSlds[c * LSTR + h4 * 4] = v;
  }
  __syncthreads();

  const v8f vzero = {0.f, 0.f, 0.f, 0.f, 0.f, 0.f, 0.f, 0.f};
  v8f acc[2][6];
  #pragma unroll
  for (int i = 0; i < 2; ++i)
    #pragma unroll
    for (int j = 0; j < 6; ++j) acc[i][j] = vzero;

  // b64 fragment addresses: lanes 0-15 get (K0,K1), lanes 16-31 get (K2,K3),
  // matching the ISA 16x4 f32 A/B VGPR layout. Conflict-free with LSTR=132.
  const int arow0 = ((mgrp * 2 + 0) * 16 + l16) * LSTR + 2 * half;
  const int arow1 = ((mgrp * 2 + 1) * 16 + l16) * LSTR + 2 * half;
  const int brow  = (ngrp * 6 * 16 + l16) * LSTR + 2 * half;

  #pragma unroll 4
  for (int k0 = 0; k0 < HEAD_DIM; k0 += 4) {
    const v2f a0 = *(const v2f*)&Alds[arow0 + k0];
    const v2f a1 = *(const v2f*)&Alds[arow1 + k0];
    #pragma unroll
    for (int j = 0; j < 6; ++j) {
      const v2f bb = *(const v2f*)&BSlds[brow + j * 16 * LSTR + k0];
      acc[0][j] = __builtin_amdgcn_wmma_f32_16x16x4_f32(
          false, a0, false, bb, (short)0, acc[0][j], false, false);
      acc[1][j] = __builtin_amdgcn_wmma_f32_16x16x4_f32(
          false, a1, false, bb, (short)0, acc[1][j], false, false);
    }
  }

  // N-group-1 waves own N'-tiles 10,11 (= bd f-tiles, j = 4,5): spill them so
  // the relative-shift gather (cross-lane) becomes a per-lane ds_load.
  // C/D layout: VGPR r -> row M = r (lanes 0-15) / r+8 (lanes 16-31), col = l16.
  if (ngrp == 1) {
    #pragma unroll
    for (int i = 0; i < 2; ++i) {
      #pragma unroll
      for (int j = 4; j < 6; ++j) {
        const int f = (j - 4) * 16 + l16;
        if (f < SPAN) {
          #pragma unroll
          for (int r = 0; r < 8; ++r) {
            const int w = (mgrp * 2 + i) * 16 + r + 8 * half;
            BDlds[w * BDSTR + f] = acc[i][j][r];
          }
        }
      }
    }
  }
  __syncthreads();

  // Epilogue: out[w,c] = ac[w,c] + (0 <= c-w < 24 ? bd[w,c-w] : 0)
  float* const o = out + (size_t)blk * ((size_t)W_ * C_);
  #pragma unroll
  for (int i = 0; i < 2; ++i) {
    #pragma unroll
    for (int j = 0; j < 6; ++j) {
      const int t = ngrp * 6 + j;          // N'-tile index
      if (t < 10) {                        // C-tiles only (10,11 are bd)
        const int c = t * 16 + l16;
        #pragma unroll
        for (int r = 0; r < 8; ++r) {
          const int w = (mgrp * 2 + i) * 16 + r + 8 * half;
          float v = acc[i][j][r];
          const int d = c - w;
          if ((unsigned)d < (unsigned)SPAN) v += BDlds[w * BDSTR + d];
          if (c < C_) o[(size_t)w * C_ + c] = v;
        }
      }
    }
  }
}

extern "C" void kernel_launch(void* const* d_in, const int* in_sizes, int n_in,
                              void* d_out, int out_size, void* d_ws, size_t ws_size,
                              hipStream_t stream) {
  const float* queries  = (const float*)d_in[0];
  const float* keys     = (const float*)d_in[1];
  const float* pos_proj = (const float*)d_in[2];
  float* out  = (float*)d_out;
  float* semb = (float*)d_ws;  // NUM_HEADS * FPAD * HEAD_DIM floats = 128 KB

  semb_kernel<<<4, 256, 0, stream>>>(pos_proj, semb);
  attn_kernel<<<B_ * NUM_HEADS * U_, 256, 0, stream>>>(queries, keys, semb, out);
}